// PredictionPipeline_71476845740653
// MI455X (gfx1250) — compile-verified
//
#include <hip/hip_runtime.h>
#include <hip/hip_bf16.h>
#include <cstdint>

#define W      2048
#define H      2048
#define HC     5                 // heatmap channels
#define HW     (W * H)
#define NBLK   (HC * H)          // one block per (channel,row) = 10240
#define MAXD   4096
#define TPB    256
#define PPT    8                 // pixels per thread = W / TPB
#define THRESH 0.3f

// ---- CDNA5 async global->LDS path (guarded; falls back to vector ld/st) ----
#if defined(__has_builtin)
# if __has_builtin(__builtin_amdgcn_global_load_async_to_lds_b128)
#  define USE_ASYNC_LDS 1
# endif
#endif
#ifndef USE_ASYNC_LDS
# define USE_ASYNC_LDS 0
#endif

__device__ __forceinline__ float sigm(float x) {
    return 1.0f / (1.0f + __expf(-x));
}

#if USE_ASYNC_LDS
// Exact pointee type per hipcc diagnostic: int __attribute__((vector_size(16)))
typedef int v4i_t __attribute__((vector_size(16)));
typedef __attribute__((address_space(1))) v4i_t* gptr_v4i;
typedef __attribute__((address_space(3))) v4i_t* lptr_v4i;
#endif

// copy 16 bytes global -> LDS
__device__ __forceinline__ void cp16_g2l(const float* g, float* l) {
#if USE_ASYNC_LDS
    __builtin_amdgcn_global_load_async_to_lds_b128(
        (gptr_v4i)(uintptr_t)g,
        (lptr_v4i)(uint32_t)(uintptr_t)l,
        0, 0);
#else
    *(float4*)l = *(const float4*)g;
#endif
}

__device__ __forceinline__ void stage_row(const float* gsrc, float* ldst, int t) {
    // 2048 floats = 512 x float4; 256 threads -> 2 per thread
#pragma unroll
    for (int k = 0; k < 2; ++k) {
        const int i4 = k * TPB + t;
        cp16_g2l(gsrc + i4 * 4, ldst + i4 * 4);
    }
}

// ---------------------------------------------------------------------------
// Kernel 1: detect peaks. Block b = c*H + y. Stages 3 rows (with halo) into
// LDS via async copies, sigmoid+threshold in place, 3x3 max peak test.
// Emits: per-block count + 8-bit flag mask per thread (bit k = pixel t*8+k).
// ---------------------------------------------------------------------------
__global__ void __launch_bounds__(TPB)
detect_kernel(const float* __restrict__ feat,
              int* __restrict__ counts,
              unsigned char* __restrict__ mask) {
    const int b = blockIdx.x;
    const int c = b / H;
    const int y = b % H;
    const int t = threadIdx.x;

    __shared__ __align__(16) float thr[3][W];   // 24 KB
    __shared__ int red[TPB];

    const float* base = feat + (size_t)c * HW;

    if (y > 0)     stage_row(base + (size_t)(y - 1) * W, thr[0], t);
                   stage_row(base + (size_t)y       * W, thr[1], t);
    if (y < H - 1) stage_row(base + (size_t)(y + 1) * W, thr[2], t);
#if USE_ASYNC_LDS
    asm volatile("s_wait_asynccnt 0" ::: "memory");
#endif
    // zero-fill missing halo rows (border: SAME pool window is clipped;
    // thresholded >= 0 so 0 is a safe identity for the max)
    if (y == 0) {
#pragma unroll
        for (int k = 0; k < PPT; ++k) thr[0][t * PPT + k] = 0.0f;
    }
    if (y == H - 1) {
#pragma unroll
        for (int k = 0; k < PPT; ++k) thr[2][t * PPT + k] = 0.0f;
    }
    __syncthreads();

    // in-place sigmoid + threshold on the valid rows (each element owned by
    // exactly one thread -> no race)
    const int r0 = (y > 0) ? 0 : 1;
    const int r1 = (y < H - 1) ? 2 : 1;
    for (int r = r0; r <= r1; ++r) {
#pragma unroll
        for (int k = 0; k < PPT; ++k) {
            const int i = t * PPT + k;
            const float s = sigm(thr[r][i]);
            thr[r][i] = (s > THRESH) ? s : 0.0f;
        }
    }
    __syncthreads();

    // 3x3 peak test
    int flags = 0;
#pragma unroll
    for (int k = 0; k < PPT; ++k) {
        const int x = t * PPT + k;
        const float v = thr[1][x];
        if (v > 0.0f) {
            const int xl = (x > 0) ? x - 1 : x;
            const int xr = (x < W - 1) ? x + 1 : x;
            float m = v;
#pragma unroll
            for (int r = 0; r < 3; ++r) {
                m = fmaxf(m, thr[r][xl]);
                m = fmaxf(m, thr[r][x]);
                m = fmaxf(m, thr[r][xr]);
            }
            if (v == m) flags |= (1 << k);
        }
    }
    mask[(size_t)b * TPB + t] = (unsigned char)flags;

    // block reduce of popcounts
    red[t] = __popc(flags);
    __syncthreads();
    for (int d = TPB / 2; d > 0; d >>= 1) {
        if (t < d) red[t] += red[t + d];
        __syncthreads();
    }
    if (t == 0) counts[b] = red[0];
}

// ---------------------------------------------------------------------------
// Kernel 2: exclusive scan of the 10240 block counts (single block).
// offsets[NBLK] = grand total (= n_valid).
// ---------------------------------------------------------------------------
__global__ void __launch_bounds__(1024)
scan_kernel(const int* __restrict__ counts, int* __restrict__ offsets) {
    __shared__ int s[1024];
    __shared__ int sbase;
    const int t = threadIdx.x;
    if (t == 0) sbase = 0;
    __syncthreads();
    for (int chunk = 0; chunk < NBLK / 1024; ++chunk) {
        const int i = chunk * 1024 + t;
        const int v = counts[i];
        s[t] = v;
        __syncthreads();
        for (int d = 1; d < 1024; d <<= 1) {
            const int x = (t >= d) ? s[t - d] : 0;
            __syncthreads();
            s[t] += x;
            __syncthreads();
        }
        offsets[i] = sbase + s[t] - v;      // exclusive
        __syncthreads();
        if (t == 1023) sbase += s[1023];
        __syncthreads();
    }
    if (t == 0) offsets[NBLK] = sbase;
}

// ---------------------------------------------------------------------------
// Kernel 3: ordered emission of (c,y,x) triples from the flag bitmask.
// Reads only the 2.6 MB mask (L2-resident). Preserves row-major order.
// ---------------------------------------------------------------------------
__global__ void __launch_bounds__(TPB)
emit_kernel(const unsigned char* __restrict__ mask,
            const int* __restrict__ offsets,
            int* __restrict__ det) {
    const int b = blockIdx.x;
    const int c = b / H;
    const int y = b % H;
    const int t = threadIdx.x;

    const int flags = mask[(size_t)b * TPB + t];
    const int cnt = __popc(flags);

    __shared__ int sc[TPB];
    sc[t] = cnt;
    __syncthreads();
    for (int d = 1; d < TPB; d <<= 1) {
        const int v = (t >= d) ? sc[t - d] : 0;
        __syncthreads();
        sc[t] += v;
        __syncthreads();
    }
    int g = offsets[b] + (sc[t] - cnt);     // global ordered position
#pragma unroll
    for (int k = 0; k < PPT; ++k) {
        if ((flags >> k) & 1) {
            if (g < MAXD) {
                det[3 * g + 0] = c;
                det[3 * g + 1] = y;
                det[3 * g + 2] = t * PPT + k;
            }
            ++g;
        }
    }
}

// ---------------------------------------------------------------------------
// Kernel 4: gather + finalize all five outputs (4096 slots, zero-fill tail).
// d_out layout: probs[4096] | coords[4096*3] | size_w[4096] | size_h[4096]
//               | side[4096]   (total 28672 floats)
// ---------------------------------------------------------------------------
__global__ void __launch_bounds__(TPB)
gather_kernel(const float* __restrict__ feat,
              const int* __restrict__ offsets,
              const int* __restrict__ det,
              float* __restrict__ out) {
    const int i = blockIdx.x * blockDim.x + threadIdx.x;
    if (i >= MAXD) return;
    const int total = offsets[NBLK];

    float p = 0.0f, fc = 0.0f, fy = 0.0f, fx = 0.0f, sz = 0.0f, sd = 0.0f;
    if (i < total) {
        const int c = det[3 * i + 0];
        const int y = det[3 * i + 1];
        const int x = det[3 * i + 2];
        fc = (float)c; fy = (float)y; fx = (float)x;
        const size_t pix = (size_t)y * W + x;
        float m = -1.0f;
#pragma unroll
        for (int ch = 0; ch < HC; ++ch)
            m = fmaxf(m, sigm(feat[(size_t)ch * HW + pix]));
        p  = m;                                   // probs_full[y,x] * valid(=1)
        sd = sigm(feat[(size_t)5 * HW + pix]);    // side_act
        sz = sigm(feat[(size_t)6 * HW + pix]);    // sizes_act
    }
    out[i]                 = p;
    out[MAXD + 3 * i + 0]  = fc;
    out[MAXD + 3 * i + 1]  = fy;
    out[MAXD + 3 * i + 2]  = fx;
    out[4 * MAXD + i]      = sz;  // size_width
    out[5 * MAXD + i]      = sz;  // size_height
    out[6 * MAXD + i]      = sd;  // side_sel
}

extern "C" void kernel_launch(void* const* d_in, const int* in_sizes, int n_in,
                              void* d_out, int out_size, void* d_ws, size_t ws_size,
                              hipStream_t stream) {
    const float* feat = (const float*)d_in[0];   // (1,7,2048,2048) fp32
    float* out = (float*)d_out;

    // workspace: counts | offsets | det triples | flag bitmask (~2.75 MB)
    int* counts  = (int*)d_ws;
    int* offsets = counts + NBLK;
    int* det     = offsets + NBLK + 1;
    unsigned char* mask = (unsigned char*)(det + 3 * MAXD);

    detect_kernel<<<NBLK, TPB, 0, stream>>>(feat, counts, mask);
    scan_kernel  <<<1, 1024, 0, stream>>>(counts, offsets);
    emit_kernel  <<<NBLK, TPB, 0, stream>>>(mask, offsets, det);
    gather_kernel<<<(MAXD + TPB - 1) / TPB, TPB, 0, stream>>>(feat, offsets, det, out);
}